// DeepHitLoss_23201413333493
// MI455X (gfx1250) — compile-verified
//
#include <hip/hip_runtime.h>
#include <math.h>

#ifndef __has_builtin
#define __has_builtin(x) 0
#endif

#if __has_builtin(__builtin_amdgcn_global_load_async_to_lds_b128)
#define ASYNC_LDS_B128 1
#else
#define ASYNC_LDS_B128 0
#endif
#if __has_builtin(__builtin_amdgcn_global_load_async_to_lds_b64)
#define ASYNC_LDS_B64 1
#else
#define ASYNC_LDS_B64 0
#endif

#define B_N 8192
#define T_N 128
#define THREADS 256
#define IPB 32                 // i-rows per block in pairwise kernel
#define IPW (IPB / 8)          // i-rows per wave (8 wave32 per block)
#define NBLK (B_N / IPB)       // 256 pairwise blocks
#define ALPHA 0.5f
#define EPS 1e-8f

typedef int v2i __attribute__((vector_size(8)));
typedef int v4i __attribute__((vector_size(16)));
typedef float f32x4 __attribute__((vector_size(16)));

// ---- CDNA5 async global -> LDS staging (ASYNCcnt path) ------------------
__device__ __forceinline__ void wait_async_zero() {
#if __has_builtin(__builtin_amdgcn_s_wait_asynccnt)
  __builtin_amdgcn_s_wait_asynccnt(0);
#else
  asm volatile("s_wait_asynccnt 0" ::: "memory");
#endif
}

// copy 16 bytes (lane-granular) from global to LDS
__device__ __forceinline__ void cp16_to_lds(const float* gsrc, float* ldst) {
#if ASYNC_LDS_B128
  __builtin_amdgcn_global_load_async_to_lds_b128(
      (__attribute__((address_space(1))) v4i*)gsrc,
      (__attribute__((address_space(3))) v4i*)ldst,
      0, 0);
#elif ASYNC_LDS_B64
  __builtin_amdgcn_global_load_async_to_lds_b64(
      (__attribute__((address_space(1))) v2i*)gsrc,
      (__attribute__((address_space(3))) v2i*)ldst,
      0, 0);
  __builtin_amdgcn_global_load_async_to_lds_b64(
      (__attribute__((address_space(1))) v2i*)(gsrc + 2),
      (__attribute__((address_space(3))) v2i*)(ldst + 2),
      0, 0);
#else
  *(f32x4*)ldst = *(const f32x4*)gsrc;   // fallback: vmem load + ds store
#endif
}

// ---- K1: max(durations) -------------------------------------------------
__global__ __launch_bounds__(THREADS) void dh_max_kernel(
    const float* __restrict__ targets, float* __restrict__ ws_max) {
  __shared__ float sm[THREADS];
  const int tid = threadIdx.x;
  float m = -1e30f;
  for (int i = tid; i < B_N; i += THREADS) m = fmaxf(m, targets[2 * i]);
  sm[tid] = m;
  __syncthreads();
  for (int off = THREADS / 2; off > 0; off >>= 1) {
    if (tid < off) sm[tid] = fmaxf(sm[tid], sm[tid + off]);
    __syncthreads();
  }
  if (tid == 0) ws_max[0] = sm[0];
}

// ---- K2: per-sample t_i, g_i, lik_i; deinterleave durations -------------
__global__ __launch_bounds__(THREADS) void dh_prep_kernel(
    const float* __restrict__ preds, const float* __restrict__ targets,
    const float* __restrict__ ws_max,
    int* __restrict__ t_arr, float* __restrict__ g_arr,
    float* __restrict__ lik_arr, float* __restrict__ dur_arr) {
  const int i = blockIdx.x * THREADS + threadIdx.x;
  const float maxd = ws_max[0];
  const float d = targets[2 * i];
  const float e = targets[2 * i + 1];
  int t = (int)((d / maxd) * (float)(T_N - 1));   // matches jnp truncation
  t = min(max(t, 0), T_N - 1);
  const float g = preds[(size_t)i * T_N + t];
  t_arr[i] = t;
  g_arr[i] = g;
  dur_arr[i] = d;
  lik_arr[i] = -logf(g + EPS) * e;                // EVENT_W == CENS_W == 1
}

// ---- K3: transpose preds (B x T) -> PT (T x B), LDS 32x32 tiles ---------
__global__ __launch_bounds__(THREADS) void dh_transpose_kernel(
    const float* __restrict__ preds, float* __restrict__ pt) {
  __shared__ float tile[32][33];
  const int lx = threadIdx.x & 31;
  const int ly = threadIdx.x >> 5;                // 0..7
  const int c0 = blockIdx.x * 32;                 // T tile (grid.x = 4)
  const int r0 = blockIdx.y * 32;                 // B tile (grid.y = 256)
#pragma unroll
  for (int k = 0; k < 4; ++k)
    tile[ly + k * 8][lx] = preds[(size_t)(r0 + ly + k * 8) * T_N + c0 + lx];
  __syncthreads();
#pragma unroll
  for (int k = 0; k < 4; ++k)
    pt[(size_t)(c0 + ly + k * 8) * B_N + r0 + lx] = tile[lx][ly + k * 8];
}

// ---- K4: pairwise ranking term ------------------------------------------
// grid = NBLK blocks x 256 threads (8 wave32). Block handles IPB i-rows.
// durations[0..B) staged once into LDS via async copies; PT rows streamed
// from L2 with coalesced float4 loads. Rows with event==0 skipped entirely.
// 4 independent accumulator chains per lane keep the FP-add pipe full.
__global__ __launch_bounds__(THREADS) void dh_pair_kernel(
    const float* __restrict__ pt, const float* __restrict__ targets,
    const int* __restrict__ t_arr, const float* __restrict__ g_arr,
    const float* __restrict__ dur_arr,
    float* __restrict__ rpart, unsigned* __restrict__ cpart) {
  __shared__ __align__(16) float sdur[B_N];       // 32 KB
  __shared__ float s_acc[IPB];
  __shared__ unsigned s_cnt[IPB];

  const int tid = threadIdx.x;
  const int wave = tid >> 5;
  const int lane = tid & 31;
  const int i0 = blockIdx.x * IPB;

  // stage all 8192 durations into LDS (async, 16B per lane-issue)
#pragma unroll
  for (int k = 0; k < B_N / 4 / THREADS; ++k) {   // 8 issues/thread
    const int p = (tid + k * THREADS) * 4;
    cp16_to_lds(dur_arr + p, &sdur[p]);
  }
  wait_async_zero();
  __syncthreads();

  int ti[IPW];
  float gi[IPW], di[IPW];
  int evi[IPW];
#pragma unroll
  for (int r = 0; r < IPW; ++r) {
    const int i = i0 + wave * IPW + r;
    ti[r] = t_arr[i];
    gi[r] = g_arr[i];
    di[r] = dur_arr[i];
    evi[r] = (targets[2 * i + 1] != 0.0f);
  }

  float acc[IPW];
  unsigned cnt[IPW];
#pragma unroll
  for (int r = 0; r < IPW; ++r) { acc[r] = 0.0f; cnt[r] = 0u; }

#pragma unroll
  for (int r = 0; r < IPW; ++r) {
    if (!evi[r]) continue;                         // wave-uniform skip
    const float* prow = pt + (size_t)ti[r] * B_N;
    const float g = gi[r];
    const float d0 = di[r];
    float ax = 0.0f, ay = 0.0f, az = 0.0f, aw = 0.0f;
    unsigned cx = 0u, cy = 0u, cz = 0u, cw = 0u;
#pragma unroll 2
    for (int jb = lane * 4; jb < B_N; jb += 32 * 4) {
      const float4 v  = *(const float4*)(prow + jb);
      const float4 dj = *(const float4*)(&sdur[jb]);
      if (dj.x > d0) { ax += fmaxf(v.x - g, 0.0f); cx += 1u; }
      if (dj.y > d0) { ay += fmaxf(v.y - g, 0.0f); cy += 1u; }
      if (dj.z > d0) { az += fmaxf(v.z - g, 0.0f); cz += 1u; }
      if (dj.w > d0) { aw += fmaxf(v.w - g, 0.0f); cw += 1u; }
    }
    acc[r] = (ax + ay) + (az + aw);
    cnt[r] = (cx + cy) + (cz + cw);
  }

  // wave32 reductions, then fixed-order block combine (deterministic)
#pragma unroll
  for (int r = 0; r < IPW; ++r) {
    float a = acc[r];
    unsigned c = cnt[r];
    for (int off = 16; off > 0; off >>= 1) {
      a += __shfl_down(a, off, 32);
      c += __shfl_down(c, off, 32);
    }
    if (lane == 0) { s_acc[wave * IPW + r] = a; s_cnt[wave * IPW + r] = c; }
  }
  __syncthreads();
  if (tid == 0) {
    float a = 0.0f;
    unsigned c = 0u;
    for (int k = 0; k < IPB; ++k) { a += s_acc[k]; c += s_cnt[k]; }
    rpart[blockIdx.x] = a;
    cpart[blockIdx.x] = c;
  }
}

// ---- K5: finalize --------------------------------------------------------
__global__ __launch_bounds__(THREADS) void dh_final_kernel(
    const float* __restrict__ lik_arr, const float* __restrict__ rpart,
    const unsigned* __restrict__ cpart, float* __restrict__ out) {
  __shared__ float sf[THREADS];
  __shared__ unsigned su[THREADS];
  const int tid = threadIdx.x;

  float s = 0.0f;
  for (int k = tid; k < B_N; k += THREADS) s += lik_arr[k];
  sf[tid] = s;
  __syncthreads();
  for (int off = THREADS / 2; off > 0; off >>= 1) {
    if (tid < off) sf[tid] += sf[tid + off];
    __syncthreads();
  }
  const float lik_sum = sf[0];
  __syncthreads();

  sf[tid] = rpart[tid];                            // NBLK == THREADS == 256
  su[tid] = cpart[tid];
  __syncthreads();
  for (int off = THREADS / 2; off > 0; off >>= 1) {
    if (tid < off) { sf[tid] += sf[tid + off]; su[tid] += su[tid + off]; }
    __syncthreads();
  }
  if (tid == 0) {
    const float rank_sum = sf[0];
    const unsigned count = su[0];
    const float rank = (count > 0u) ? rank_sum / (float)count : 0.0f;
    out[0] = ALPHA * (lik_sum / (float)B_N) + (1.0f - ALPHA) * rank;
  }
}

// ---- host-side launch ----------------------------------------------------
extern "C" void kernel_launch(void* const* d_in, const int* in_sizes, int n_in,
                              void* d_out, int out_size, void* d_ws, size_t ws_size,
                              hipStream_t stream) {
  const float* preds = (const float*)d_in[0];     // (8192, 128) f32
  const float* targets = (const float*)d_in[1];   // (8192, 2)   f32
  float* out = (float*)d_out;                     // scalar f32

  char* ws = (char*)d_ws;
  float* ws_max = (float*)(ws + 0);               // 4 B (padded to 256)
  float* pt     = (float*)(ws + 256);             // 128*8192 f32 = 4 MB
  char* cur = ws + 256 + (size_t)T_N * B_N * sizeof(float);
  int*   t_arr   = (int*)cur;    cur += B_N * sizeof(int);
  float* g_arr   = (float*)cur;  cur += B_N * sizeof(float);
  float* dur_arr = (float*)cur;  cur += B_N * sizeof(float);
  float* lik_arr = (float*)cur;  cur += B_N * sizeof(float);
  float* rpart   = (float*)cur;  cur += NBLK * sizeof(float);
  unsigned* cpart = (unsigned*)cur;

  dh_max_kernel<<<1, THREADS, 0, stream>>>(targets, ws_max);
  dh_prep_kernel<<<B_N / THREADS, THREADS, 0, stream>>>(
      preds, targets, ws_max, t_arr, g_arr, lik_arr, dur_arr);
  dh_transpose_kernel<<<dim3(T_N / 32, B_N / 32), THREADS, 0, stream>>>(preds, pt);
  dh_pair_kernel<<<NBLK, THREADS, 0, stream>>>(
      pt, targets, t_arr, g_arr, dur_arr, rpart, cpart);
  dh_final_kernel<<<1, THREADS, 0, stream>>>(lik_arr, rpart, cpart, out);
}